// TopKGate_90864328114994
// MI455X (gfx1250) — compile-verified
//
#include <hip/hip_runtime.h>

// MoE top-2 gating for MI455X (gfx1250, wave32).
// logits = h[16384,4096] @ W[64,4096]^T via V_WMMA_F32_16X16X4_F32,
// then softmax/top-2/renorm epilogue through LDS.

typedef __attribute__((ext_vector_type(2))) float v2f;
typedef __attribute__((ext_vector_type(8))) float v8f;

constexpr int TOKENS         = 16384;
constexpr int DIM            = 4096;
constexpr int NEXP           = 64;
constexpr int WAVES_PER_BLK  = 4;
constexpr int TOK_PER_WAVE   = 16;
constexpr int TOK_PER_BLK    = WAVES_PER_BLK * TOK_PER_WAVE;  // 64
constexpr float EPS          = 1e-9f;

__global__ __launch_bounds__(WAVES_PER_BLK * 32)
void topk_gate_kernel(const float* __restrict__ h,
                      const float* __restrict__ W,
                      float* __restrict__ w_out,
                      int*   __restrict__ i_out)
{
    __shared__ float slds[TOK_PER_BLK * NEXP];   // 16 KB logits staging

    const int lane  = threadIdx.x & 31;
    const int wave  = threadIdx.x >> 5;
    const int lrow  = lane & 15;                 // M (A/C) or N (B) index
    const int khalf = (lane >> 4) << 1;          // lanes 0-15: K+0/1, 16-31: K+2/3

    const int tokBase = blockIdx.x * TOK_PER_BLK + wave * TOK_PER_WAVE;

    // A fragment source: one token row per lane (16 rows, both lane halves).
    const float* aptr  = h + (size_t)(tokBase + lrow) * DIM + khalf;
    // B fragment sources: 4 expert tiles of 16 experts each (W rows = experts).
    const float* bptr0 = W + (size_t)( 0 + lrow) * DIM + khalf;
    const float* bptr1 = W + (size_t)(16 + lrow) * DIM + khalf;
    const float* bptr2 = W + (size_t)(32 + lrow) * DIM + khalf;
    const float* bptr3 = W + (size_t)(48 + lrow) * DIM + khalf;

    v8f acc0 = {}, acc1 = {}, acc2 = {}, acc3 = {};

    #pragma unroll 4
    for (int k = 0; k < DIM; k += 4) {
        v2f a  = *(const v2f*)(aptr  + k);
        v2f b0 = *(const v2f*)(bptr0 + k);
        v2f b1 = *(const v2f*)(bptr1 + k);
        v2f b2 = *(const v2f*)(bptr2 + k);
        v2f b3 = *(const v2f*)(bptr3 + k);
        // (neg_a, A, neg_b, B, c_mod, C, reuse_a, reuse_b)
        acc0 = __builtin_amdgcn_wmma_f32_16x16x4_f32(false, a, false, b0, (short)0, acc0, false, false);
        acc1 = __builtin_amdgcn_wmma_f32_16x16x4_f32(false, a, false, b1, (short)0, acc1, false, false);
        acc2 = __builtin_amdgcn_wmma_f32_16x16x4_f32(false, a, false, b2, (short)0, acc2, false, false);
        acc3 = __builtin_amdgcn_wmma_f32_16x16x4_f32(false, a, false, b3, (short)0, acc3, false, false);
    }

    // C/D layout: VGPR v, lanes 0-15 -> M=v, N=lane; lanes 16-31 -> M=v+8, N=lane-16.
    const int trow = (lane >> 4) * 8;
    float* wlds = slds + (wave * TOK_PER_WAVE) * NEXP;
    #pragma unroll
    for (int v = 0; v < 8; ++v) {
        wlds[(trow + v) * NEXP +  0 + lrow] = acc0[v];
        wlds[(trow + v) * NEXP + 16 + lrow] = acc1[v];
        wlds[(trow + v) * NEXP + 32 + lrow] = acc2[v];
        wlds[(trow + v) * NEXP + 48 + lrow] = acc3[v];
    }

    __syncthreads();

    // One lane per token: top-2 (lax.top_k tie semantics: strict >, lowest
    // index first), full softmax denominator, renormalized gathered weights.
    const int t = threadIdx.x;
    if (t < TOK_PER_BLK) {
        const float* row = slds + t * NEXP;
        float m1 = -__builtin_huge_valf(); int i1 = 0;
        float m2 = -__builtin_huge_valf(); int i2 = 0;
        #pragma unroll 8
        for (int j = 0; j < NEXP; ++j) {
            float v = row[j];
            if (v > m1)      { m2 = m1; i2 = i1; m1 = v; i1 = j; }
            else if (v > m2) { m2 = v;  i2 = j; }
        }
        float Z = 0.0f;
        #pragma unroll 8
        for (int j = 0; j < NEXP; ++j) Z += expf(row[j] - m1);
        float g1  = 1.0f / Z;             // exp(m1 - m1) / Z
        float g2  = expf(m2 - m1) / Z;
        float inv = 1.0f / (g1 + g2 + EPS);
        const int gtok = blockIdx.x * TOK_PER_BLK + t;
        w_out[gtok * 2 + 0] = g1 * inv;
        w_out[gtok * 2 + 1] = g2 * inv;
        i_out[gtok * 2 + 0] = i1;
        i_out[gtok * 2 + 1] = i2;
    }
}

extern "C" void kernel_launch(void* const* d_in, const int* in_sizes, int n_in,
                              void* d_out, int out_size, void* d_ws, size_t ws_size,
                              hipStream_t stream)
{
    const float* h = (const float*)d_in[0];   // [16384, 4096]
    const float* W = (const float*)d_in[1];   // [64, 4096]
    float* w_out = (float*)d_out;                         // [16384, 2] f32
    int*   i_out = (int*)d_out + (size_t)TOKENS * 2;      // [16384, 2] i32 (bit-cast region)

    dim3 grid(TOKENS / TOK_PER_BLK);          // 256 blocks
    dim3 block(WAVES_PER_BLK * 32);           // 128 threads = 4 waves
    hipLaunchKernelGGL(topk_gate_kernel, grid, block, 0, stream,
                       h, W, w_out, i_out);
}